// MyBertPooler_Stochastic_76192719831707
// MI455X (gfx1250) — compile-verified
//
#include <hip/hip_runtime.h>
#include <hip/hip_bf16.h>

typedef __attribute__((ext_vector_type(2))) float v2f;
typedef __attribute__((ext_vector_type(8))) float v8f;

#define B_DIM 8
#define S_DIM 4096
#define H_DIM 1024
#define K_SAMPLE 8

// ---------------------------------------------------------------------------
// Kernel 1: per-row Gumbel top-8 pooling.
// One wave32 per row r in [0, B*H). Row r corresponds to (b = r/H, h = r%H);
// t[r, s] = relu(hidden_states[b, s, h]), score[s] = log(t) + gumbel[r, s].
// pooled[r] = sum of t over the top-8 scores (0 if the whole row is zero).
// Lanes stride s (gumbel coalesced, non-temporal: single use); hs loads get
// cache-line reuse across the 32 blocks sharing each (b,s) line via L2.
// ---------------------------------------------------------------------------
__global__ __launch_bounds__(32) void topk_pool_kernel(
    const float* __restrict__ hs,      // [B, S, H]
    const float* __restrict__ gumbel,  // [B*H, S]
    float* __restrict__ pooled)        // [B*H]
{
    const int r    = blockIdx.x;          // row 0..8191
    const int lane = threadIdx.x;         // 0..31
    const int b    = r >> 10;             // r / H_DIM
    const int h    = r & (H_DIM - 1);     // r % H_DIM

    const float* grow = gumbel + (size_t)r * S_DIM;

    float s8[K_SAMPLE];                   // scores of current top-8
    float v8[K_SAMPLE];                   // matching values
#pragma unroll
    for (int i = 0; i < K_SAMPLE; ++i) { s8[i] = -__builtin_inff(); v8[i] = 0.0f; }
    float smin = -__builtin_inff();       // min score currently held
    int   imin = 0;
    float psum = 0.0f;                    // partial row sum (for zero-row mask)

    for (int s = lane; s < S_DIM; s += 32) {
        float x = hs[((size_t)b * S_DIM + s) * H_DIM + h];
        x = x > 0.0f ? x : 0.0f;          // relu
        psum += x;
        // logf(0) = -inf -> zero entries are never selected over real ones.
        float g  = __builtin_nontemporal_load(grow + s);
        float sc = logf(x) + g;
        if (sc > smin) {
            s8[imin] = sc;
            v8[imin] = x;
            // re-find the min slot
            smin = s8[0]; imin = 0;
#pragma unroll
            for (int i = 1; i < K_SAMPLE; ++i) {
                if (s8[i] < smin) { smin = s8[i]; imin = i; }
            }
        }
    }

    __shared__ float shS[32 * K_SAMPLE];
    __shared__ float shV[32 * K_SAMPLE];
    __shared__ float shP[32];
#pragma unroll
    for (int i = 0; i < K_SAMPLE; ++i) {
        shS[lane * K_SAMPLE + i] = s8[i];
        shV[lane * K_SAMPLE + i] = v8[i];
    }
    shP[lane] = psum;
    __syncthreads();

    if (lane == 0) {
        float rowsum = 0.0f;
        for (int i = 0; i < 32; ++i) rowsum += shP[i];
        float acc = 0.0f;
        for (int t = 0; t < K_SAMPLE; ++t) {
            int   bi = -1;
            float bs = -__builtin_inff();
            for (int i = 0; i < 32 * K_SAMPLE; ++i) {
                if (shS[i] > bs) { bs = shS[i]; bi = i; }
            }
            if (bi >= 0) {
                acc += shV[bi];
                shS[bi] = -__builtin_inff();
            }
        }
        // Zero row: reference samples uniformly from (t+1) then subtracts the
        // +1 back from each of the k gathered values -> exact result 0.
        pooled[r] = (rowsum > 0.0f) ? acc : 0.0f;
    }
}

// ---------------------------------------------------------------------------
// Kernel 2: out = tanh(pooled @ W^T + bias) via V_WMMA_F32_16X16X4_F32.
// D(16x16) = A(16x4) * B(4x16) + C, K looped over H in steps of 4.
// A: M = batch row (8 real rows; rows 8..15 padded via branchless 0/1 mask so
//    the loop body has NO exec-mask manipulation).
//    lanes 0-15 hold M=lane, K={k0,k0+1}; lanes 16-31 hold M=lane-16,
//    K={k0+2,k0+3}  (ISA 7.12.2, 32-bit A 16x4 layout).
// B: B[k][n] = W^T[k][n] = W[n][k]; same K split across lane halves, N=lane&15.
// D: VGPR i -> lanes 0-15: M=i (valid, stored); lanes 16-31: M=i+8 (pad).
// ---------------------------------------------------------------------------
__global__ __launch_bounds__(32) void gemm_tanh_kernel(
    const float* __restrict__ pooled,  // [B, H]  (B=8 rows, contiguous)
    const float* __restrict__ W,       // [H, H]
    const float* __restrict__ bias,    // [H]
    float* __restrict__ out)           // [B, H]
{
    const int lane   = threadIdx.x;        // 0..31
    const int n_tile = blockIdx.x * 16;    // output-column tile
    const int half   = lane >> 4;          // 0: K pair {0,1}, 1: K pair {2,3}
    const int idx16  = lane & 15;          // M (for A) / N (for B,D)
    const int j      = n_tile + idx16;     // W row feeding column N=idx16

    // Branchless pad handling: pad rows read row (idx16&7), then zeroed.
    const float amask = (idx16 < B_DIM) ? 1.0f : 0.0f;
    const float* pa = pooled + (size_t)(idx16 & 7) * H_DIM + half * 2;
    const float* pb = W      + (size_t)j           * H_DIM + half * 2;

    v8f c = {};                            // accumulator (C = 0)
#pragma unroll 8
    for (int k0 = 0; k0 < H_DIM; k0 += 4) {
        v2f a = *(const v2f*)(pa + k0);
        a.x *= amask;
        a.y *= amask;
        v2f bm = *(const v2f*)(pb + k0);
        // 8 args: (neg_a, A, neg_b, B, c_mod, C, reuse_a, reuse_b)
        c = __builtin_amdgcn_wmma_f32_16x16x4_f32(
                false, a, false, bm, (short)0, c, false, false);
    }

    if (half == 0) {                       // lanes 0..15 hold M = 0..7 rows
        const float bj = bias[j];
#pragma unroll
        for (int i = 0; i < B_DIM; ++i) {
            out[(size_t)i * H_DIM + j] = tanhf(c[i] + bj);
        }
    }
}

// ---------------------------------------------------------------------------
extern "C" void kernel_launch(void* const* d_in, const int* in_sizes, int n_in,
                              void* d_out, int out_size, void* d_ws, size_t ws_size,
                              hipStream_t stream) {
    const float* hs     = (const float*)d_in[0];  // hidden_states [8,4096,1024]
    const float* gumbel = (const float*)d_in[1];  // gumbel [8192, 4096]
    const float* W      = (const float*)d_in[2];  // [1024, 1024]
    const float* bias   = (const float*)d_in[3];  // [1024]
    // d_in[4] = sample (== 8, baked in as K_SAMPLE)

    float* pooled = (float*)d_ws;                 // [8192] floats (32 KB)

    topk_pool_kernel<<<B_DIM * H_DIM, 32, 0, stream>>>(hs, gumbel, pooled);
    gemm_tanh_kernel<<<H_DIM / 16, 32, 0, stream>>>(pooled, W, bias, (float*)d_out);
}